// HeteroGNN_14113262535048
// MI455X (gfx1250) — compile-verified
//
#include <hip/hip_runtime.h>

typedef __attribute__((ext_vector_type(2))) float v2f;
typedef __attribute__((ext_vector_type(8))) float v8f;

#define N_BUS 200000
#define N_GEN 50000
#define E_BB  1600000
#define E_GB  400000
#define E_BG  400000
#define D_IN  64
#define HDIM  32
#define OUT_BUS 4
#define OUT_GEN 2

// ---------------------------------------------------------------------------
// WMMA GEMM: O[N x 32] = (accum ? O : 0) + X[N x K] @ W[K x 32], fp32 exact.
// One wave per 16-row tile (grid.x = N/16, blockDim = 32) -> EXEC always all 1s.
// Uses V_WMMA_F32_16X16X4_F32: A = 16x4 f32 (2 VGPR), B = 4x16 f32 (2 VGPR).
// A layout: lane m=lane&15 holds row m; kh=lane>>4 selects K pair {k+2kh, k+2kh+1}.
// D layout: lanes 0-15 -> rows r (VGPR idx), lanes 16-31 -> rows r+8; col = lane&15.
// ---------------------------------------------------------------------------
__global__ __launch_bounds__(32) void gemm_wmma(const float* __restrict__ X,
                                                const float* __restrict__ W,
                                                float* __restrict__ O,
                                                int K, int accum)
{
    const int lane = threadIdx.x & 31;
    const int m    = lane & 15;
    const int kh   = lane >> 4;
    const long long tile = blockIdx.x;
    const float* xr = X + (tile * 16 + m) * (long long)K;

    v8f c0 = {0.f,0.f,0.f,0.f,0.f,0.f,0.f,0.f};
    v8f c1 = {0.f,0.f,0.f,0.f,0.f,0.f,0.f,0.f};

    for (int k = 0; k < K; k += 4) {
        const int kb = k + 2 * kh;
        v2f a, b0, b1;
        a.x  = xr[kb];
        a.y  = xr[kb + 1];
        b0.x = W[(long long)kb * HDIM + m];
        b0.y = W[(long long)(kb + 1) * HDIM + m];
        b1.x = W[(long long)kb * HDIM + 16 + m];
        b1.y = W[(long long)(kb + 1) * HDIM + 16 + m];
        c0 = __builtin_amdgcn_wmma_f32_16x16x4_f32(false, a, false, b0, (short)0, c0, false, false);
        c1 = __builtin_amdgcn_wmma_f32_16x16x4_f32(false, a, false, b1, (short)0, c1, false, false);
    }

    const long long rbase = tile * 16 + kh * 8;
    if (accum) {
#pragma unroll
        for (int r = 0; r < 8; ++r) {
            O[(rbase + r) * HDIM + m]      += c0[r];
            O[(rbase + r) * HDIM + 16 + m] += c1[r];
        }
    } else {
#pragma unroll
        for (int r = 0; r < 8; ++r) {
            O[(rbase + r) * HDIM + m]      = c0[r];
            O[(rbase + r) * HDIM + 16 + m] = c1[r];
        }
    }
}

// Degree counts (float, so the divide path is branch-free later).
__global__ void count_deg(const int* __restrict__ dst, int nE, float* __restrict__ cnt)
{
    int gid = blockIdx.x * blockDim.x + threadIdx.x;
    if (gid < nE) atomicAdd(&cnt[dst[gid]], 1.0f);
}

// Scatter-add of projected 32-float messages.
// 8 lanes per edge; each lane moves one float4 (b128 load + 4 f32 atomics).
// Same bytes as scalar version but ~2.3x fewer vmem issue slots.
__global__ void scatter_h(const float* __restrict__ y, const int* __restrict__ src,
                          const int* __restrict__ dst, int nE, float* __restrict__ agg)
{
    long long gid = (long long)blockIdx.x * blockDim.x + threadIdx.x;
    if (gid >= (long long)nE * 8) return;
    int e  = (int)(gid >> 3);
    int f4 = (int)(gid & 7) * 4;
    const float4 v = *(const float4*)(y + (long long)src[e] * HDIM + f4);
    float* a = agg + (long long)dst[e] * HDIM + f4;
    atomicAdd(a + 0, v.x);
    atomicAdd(a + 1, v.y);
    atomicAdd(a + 2, v.z);
    atomicAdd(a + 3, v.w);
}

// bus dst: two relations (bb, gb) with separate means + biases, plus root; ReLU.
// One thread per float4 (contiguous streams -> b128 traffic).
__global__ void finalize_bus(const float* __restrict__ agg_bb, const float* __restrict__ agg_gb,
                             const float* __restrict__ cnt_bb, const float* __restrict__ cnt_gb,
                             const float* __restrict__ bl_bb, const float* __restrict__ bl_gb,
                             const float* __restrict__ rt, float* __restrict__ h)
{
    long long gid = (long long)blockIdx.x * blockDim.x + threadIdx.x;   // float4 index
    if (gid >= (long long)N_BUS * (HDIM / 4)) return;
    int n  = (int)(gid >> 3);
    int f4 = (int)(gid & 7) * 4;
    float ibb = 1.0f / fmaxf(cnt_bb[n], 1.0f);
    float igb = 1.0f / fmaxf(cnt_gb[n], 1.0f);
    float4 abb = ((const float4*)agg_bb)[gid];
    float4 agb = ((const float4*)agg_gb)[gid];
    float4 r   = ((const float4*)rt)[gid];
    float4 o;
    o.x = fmaxf(abb.x * ibb + bl_bb[f4 + 0] + agb.x * igb + bl_gb[f4 + 0] + r.x, 0.f);
    o.y = fmaxf(abb.y * ibb + bl_bb[f4 + 1] + agb.y * igb + bl_gb[f4 + 1] + r.y, 0.f);
    o.z = fmaxf(abb.z * ibb + bl_bb[f4 + 2] + agb.z * igb + bl_gb[f4 + 2] + r.z, 0.f);
    o.w = fmaxf(abb.w * ibb + bl_bb[f4 + 3] + agb.w * igb + bl_gb[f4 + 3] + r.w, 0.f);
    ((float4*)h)[gid] = o;
}

// gen dst: single relation (bg).
__global__ void finalize_gen(const float* __restrict__ agg_bg, const float* __restrict__ cnt_bg,
                             const float* __restrict__ bl_bg, const float* __restrict__ rt,
                             float* __restrict__ h)
{
    long long gid = (long long)blockIdx.x * blockDim.x + threadIdx.x;   // float4 index
    if (gid >= (long long)N_GEN * (HDIM / 4)) return;
    int n  = (int)(gid >> 3);
    int f4 = (int)(gid & 7) * 4;
    float ibg = 1.0f / fmaxf(cnt_bg[n], 1.0f);
    float4 a = ((const float4*)agg_bg)[gid];
    float4 r = ((const float4*)rt)[gid];
    float4 o;
    o.x = fmaxf(a.x * ibg + bl_bg[f4 + 0] + r.x, 0.f);
    o.y = fmaxf(a.y * ibg + bl_bg[f4 + 1] + r.y, 0.f);
    o.z = fmaxf(a.z * ibg + bl_bg[f4 + 2] + r.z, 0.f);
    o.w = fmaxf(a.w * ibg + bl_bg[f4 + 3] + r.w, 0.f);
    ((float4*)h)[gid] = o;
}

// Fold lin0 (HxH + bias) and linf (HxOUT + bias) into one HxOUT affine map
// (no ReLU between them in the reference).
__global__ void combine_head(const float* __restrict__ lin0_W, const float* __restrict__ lin0_b,
                             const float* __restrict__ linf_W, const float* __restrict__ linf_b,
                             int OUTD, float* __restrict__ Wc, float* __restrict__ bc)
{
    int t = threadIdx.x;
    for (int idx = t; idx < HDIM * OUTD; idx += blockDim.x) {
        int f = idx / OUTD, j = idx - f * OUTD;
        float s = 0.f;
        for (int k = 0; k < HDIM; ++k) s += lin0_W[f * HDIM + k] * linf_W[k * OUTD + j];
        Wc[idx] = s;
    }
    for (int j = t; j < OUTD; j += blockDim.x) {
        float s = linf_b[j];
        for (int k = 0; k < HDIM; ++k) s += lin0_b[k] * linf_W[k * OUTD + j];
        bc[j] = s;
    }
}

// One thread per node: read the 32-float row once (float4s), emit all OUTD outputs.
// h is already ReLU'd (relu idempotent in the reference).
template <int OUTD>
__global__ void head(const float* __restrict__ h, const float* __restrict__ Wc,
                     const float* __restrict__ bc, long long N, float* __restrict__ out)
{
    long long n = (long long)blockIdx.x * blockDim.x + threadIdx.x;
    if (n >= N) return;
    const float4* hr = (const float4*)(h + n * HDIM);
    float s[OUTD];
#pragma unroll
    for (int j = 0; j < OUTD; ++j) s[j] = bc[j];
#pragma unroll
    for (int q = 0; q < HDIM / 4; ++q) {
        float4 v = hr[q];
        const float* wq = Wc + q * 4 * OUTD;
#pragma unroll
        for (int j = 0; j < OUTD; ++j) {
            s[j] += v.x * wq[0 * OUTD + j] + v.y * wq[1 * OUTD + j]
                  + v.z * wq[2 * OUTD + j] + v.w * wq[3 * OUTD + j];
        }
    }
#pragma unroll
    for (int j = 0; j < OUTD; ++j) out[n * OUTD + j] = s[j];
}

// ---------------------------------------------------------------------------

extern "C" void kernel_launch(void* const* d_in, const int* in_sizes, int n_in,
                              void* d_out, int out_size, void* d_ws, size_t ws_size,
                              hipStream_t stream)
{
    (void)in_sizes; (void)n_in; (void)out_size; (void)ws_size;

    const float* x_bus  = (const float*)d_in[0];
    const float* x_gen  = (const float*)d_in[1];
    const int* src_bb   = (const int*)d_in[2];
    const int* dst_bb   = (const int*)d_in[3];
    const int* src_gb   = (const int*)d_in[4];
    const int* dst_gb   = (const int*)d_in[5];
    const int* src_bg   = (const int*)d_in[6];
    const int* dst_bg   = (const int*)d_in[7];
    // layer weights: (l1,l2) x (bb,gb,bg) x (Wl, bl, Wr)
    const float* l1_bb_Wl = (const float*)d_in[8];
    const float* l1_bb_bl = (const float*)d_in[9];
    const float* l1_bb_Wr = (const float*)d_in[10];
    const float* l1_gb_Wl = (const float*)d_in[11];
    const float* l1_gb_bl = (const float*)d_in[12];
    const float* l1_gb_Wr = (const float*)d_in[13];
    const float* l1_bg_Wl = (const float*)d_in[14];
    const float* l1_bg_bl = (const float*)d_in[15];
    const float* l1_bg_Wr = (const float*)d_in[16];
    const float* l2_bb_Wl = (const float*)d_in[17];
    const float* l2_bb_bl = (const float*)d_in[18];
    const float* l2_bb_Wr = (const float*)d_in[19];
    const float* l2_gb_Wl = (const float*)d_in[20];
    const float* l2_gb_bl = (const float*)d_in[21];
    const float* l2_gb_Wr = (const float*)d_in[22];
    const float* l2_bg_Wl = (const float*)d_in[23];
    const float* l2_bg_bl = (const float*)d_in[24];
    const float* l2_bg_Wr = (const float*)d_in[25];
    const float* lin0_bus_W = (const float*)d_in[26];
    const float* lin0_bus_b = (const float*)d_in[27];
    const float* linf_bus_W = (const float*)d_in[28];
    const float* linf_bus_b = (const float*)d_in[29];
    const float* lin0_gen_W = (const float*)d_in[30];
    const float* lin0_gen_b = (const float*)d_in[31];
    const float* linf_gen_W = (const float*)d_in[32];
    const float* linf_gen_b = (const float*)d_in[33];

    float* out_bus = (float*)d_out;
    float* out_gen = (float*)d_out + (long long)N_BUS * OUT_BUS;

    // ---- workspace layout (floats) ----
    float* w = (float*)d_ws;
    size_t o = 0;
    float* cnt_bb = w + o; o += N_BUS;
    float* cnt_gb = w + o; o += N_BUS;
    float* cnt_bg = w + o; o += N_GEN;
    // keep agg region 16B-aligned (previous fields total 450000 floats; pad to mult of 4)
    o = (o + 3) & ~(size_t)3;
    float* agg_bb = w + o; o += (size_t)N_BUS * HDIM;
    float* agg_gb = w + o; o += (size_t)N_BUS * HDIM;
    float* agg_bg = w + o; o += (size_t)N_GEN * HDIM;
    float* y_bb   = w + o; o += (size_t)N_BUS * HDIM;
    float* y_gb   = w + o; o += (size_t)N_GEN * HDIM;
    float* y_bg   = w + o; o += (size_t)N_BUS * HDIM;
    float* rt_b   = w + o; o += (size_t)N_BUS * HDIM;
    float* rt_g   = w + o; o += (size_t)N_GEN * HDIM;
    float* h_b    = w + o; o += (size_t)N_BUS * HDIM;
    float* h_g    = w + o; o += (size_t)N_GEN * HDIM;
    float* Wc_b   = w + o; o += HDIM * OUT_BUS;
    float* bc_b   = w + o; o += OUT_BUS;
    float* Wc_g   = w + o; o += HDIM * OUT_GEN;
    float* bc_g   = w + o; o += OUT_GEN;

    const size_t agg_bytes = (size_t)(2 * N_BUS + N_GEN) * HDIM * sizeof(float);
    const size_t cnt_bytes = ((char*)agg_bb - (char*)cnt_bb);

    const int TB = 256;
    const int TILES_BUS = N_BUS / 16;   // 12500, exact
    const int TILES_GEN = N_GEN / 16;   // 3125, exact

    // zero counts + layer-1 accumulators (contiguous region)
    hipMemsetAsync(cnt_bb, 0, cnt_bytes + agg_bytes, stream);

    // degree counts (reused by both layers)
    count_deg<<<(E_BB + TB - 1) / TB, TB, 0, stream>>>(dst_bb, E_BB, cnt_bb);
    count_deg<<<(E_GB + TB - 1) / TB, TB, 0, stream>>>(dst_gb, E_GB, cnt_gb);
    count_deg<<<(E_BG + TB - 1) / TB, TB, 0, stream>>>(dst_bg, E_BG, cnt_bg);

    // ---------------- Layer 1 (K = 64) ----------------
    gemm_wmma<<<TILES_BUS, 32, 0, stream>>>(x_bus, l1_bb_Wl, y_bb, D_IN, 0);
    gemm_wmma<<<TILES_GEN, 32, 0, stream>>>(x_gen, l1_gb_Wl, y_gb, D_IN, 0);
    gemm_wmma<<<TILES_BUS, 32, 0, stream>>>(x_bus, l1_bg_Wl, y_bg, D_IN, 0);
    gemm_wmma<<<TILES_BUS, 32, 0, stream>>>(x_bus, l1_bb_Wr, rt_b, D_IN, 0);
    gemm_wmma<<<TILES_BUS, 32, 0, stream>>>(x_bus, l1_gb_Wr, rt_b, D_IN, 1);
    gemm_wmma<<<TILES_GEN, 32, 0, stream>>>(x_gen, l1_bg_Wr, rt_g, D_IN, 0);

    scatter_h<<<(unsigned)(((long long)E_BB * 8 + TB - 1) / TB), TB, 0, stream>>>(y_bb, src_bb, dst_bb, E_BB, agg_bb);
    scatter_h<<<(unsigned)(((long long)E_GB * 8 + TB - 1) / TB), TB, 0, stream>>>(y_gb, src_gb, dst_gb, E_GB, agg_gb);
    scatter_h<<<(unsigned)(((long long)E_BG * 8 + TB - 1) / TB), TB, 0, stream>>>(y_bg, src_bg, dst_bg, E_BG, agg_bg);

    finalize_bus<<<(unsigned)(((long long)N_BUS * 8 + TB - 1) / TB), TB, 0, stream>>>(
        agg_bb, agg_gb, cnt_bb, cnt_gb, l1_bb_bl, l1_gb_bl, rt_b, h_b);
    finalize_gen<<<(unsigned)(((long long)N_GEN * 8 + TB - 1) / TB), TB, 0, stream>>>(
        agg_bg, cnt_bg, l1_bg_bl, rt_g, h_g);

    // ---------------- Layer 2 (K = 32) ----------------
    hipMemsetAsync(agg_bb, 0, agg_bytes, stream);

    gemm_wmma<<<TILES_BUS, 32, 0, stream>>>(h_b, l2_bb_Wl, y_bb, HDIM, 0);
    gemm_wmma<<<TILES_GEN, 32, 0, stream>>>(h_g, l2_gb_Wl, y_gb, HDIM, 0);
    gemm_wmma<<<TILES_BUS, 32, 0, stream>>>(h_b, l2_bg_Wl, y_bg, HDIM, 0);
    gemm_wmma<<<TILES_BUS, 32, 0, stream>>>(h_b, l2_bb_Wr, rt_b, HDIM, 0);
    gemm_wmma<<<TILES_BUS, 32, 0, stream>>>(h_b, l2_gb_Wr, rt_b, HDIM, 1);
    gemm_wmma<<<TILES_GEN, 32, 0, stream>>>(h_g, l2_bg_Wr, rt_g, HDIM, 0);

    scatter_h<<<(unsigned)(((long long)E_BB * 8 + TB - 1) / TB), TB, 0, stream>>>(y_bb, src_bb, dst_bb, E_BB, agg_bb);
    scatter_h<<<(unsigned)(((long long)E_GB * 8 + TB - 1) / TB), TB, 0, stream>>>(y_gb, src_gb, dst_gb, E_GB, agg_gb);
    scatter_h<<<(unsigned)(((long long)E_BG * 8 + TB - 1) / TB), TB, 0, stream>>>(y_bg, src_bg, dst_bg, E_BG, agg_bg);

    // h_b / h_g overwritten in place (not read by finalize)
    finalize_bus<<<(unsigned)(((long long)N_BUS * 8 + TB - 1) / TB), TB, 0, stream>>>(
        agg_bb, agg_gb, cnt_bb, cnt_gb, l2_bb_bl, l2_gb_bl, rt_b, h_b);
    finalize_gen<<<(unsigned)(((long long)N_GEN * 8 + TB - 1) / TB), TB, 0, stream>>>(
        agg_bg, cnt_bg, l2_bg_bl, rt_g, h_g);

    // ---------------- Heads (lin0 + linf folded) ----------------
    combine_head<<<1, 128, 0, stream>>>(lin0_bus_W, lin0_bus_b, linf_bus_W, linf_bus_b,
                                        OUT_BUS, Wc_b, bc_b);
    combine_head<<<1, 128, 0, stream>>>(lin0_gen_W, lin0_gen_b, linf_gen_W, linf_gen_b,
                                        OUT_GEN, Wc_g, bc_g);

    head<OUT_BUS><<<(N_BUS + TB - 1) / TB, TB, 0, stream>>>(h_b, Wc_b, bc_b, N_BUS, out_bus);
    head<OUT_GEN><<<(N_GEN + TB - 1) / TB, TB, 0, stream>>>(h_g, Wc_g, bc_g, N_GEN, out_gen);
}